// ResidualVectorQuantizer_48344151884178
// MI455X (gfx1250) — compile-verified
//
#include <hip/hip_runtime.h>
#include <stdint.h>

// ---------------------------------------------------------------------------
// Residual VQ forward for MI455X (gfx1250, wave32).
//  - fp32 WMMA (v_wmma_f32_16x16x4_f32) for the 24000x1024x256 score GEMMs
//  - residual held in registers in A-fragment layout across all 8 stages
//  - codebook tiles double-buffered in LDS via global_load_async_to_lds_b128
//    (inline asm; builtin signature is not source-spellable from HIP)
// ---------------------------------------------------------------------------

typedef __attribute__((ext_vector_type(2))) float v2f;
typedef __attribute__((ext_vector_type(8))) float v8f;

#define BB_    16
#define DD_    256
#define TT_    1500
#define NQ_    8
#define KBINS_ 1024
#define NROWS_ (BB_ * TT_)          // 24000
#define ROWS_PER_BLOCK_ 64          // 4 waves x 16 rows
#define NC_    32                   // codebook columns staged per step
#define NSTAGE_ (KBINS_ / NC_)      // 32
#define COLSTRIDE_ 132              // padded floats per column (128 + 4 pad) -> bank-conflict-free
#define QUANT_N_ (BB_ * DD_ * TT_)  // 6,144,000
#define CODES_OFF_ QUANT_N_
#define PEN_OFF_ (QUANT_N_ + NQ_ * NROWS_)

// Async global->LDS copy, 16 bytes per lane. LDS byte offset goes in a VGPR
// (flat shared pointers truncate to the LDS offset per the ISA aperture map).
__device__ __forceinline__ void async_load_b128(const float* src, float* dst_lds) {
  const uint32_t lds = (uint32_t)(uintptr_t)dst_lds;
  asm volatile("global_load_async_to_lds_b128 %0, %1, off"
               :
               : "v"(lds), "v"(src)
               : "memory");
}

__device__ __forceinline__ void wait_asynccnt0() {
#if __has_builtin(__builtin_amdgcn_s_wait_asynccnt)
  __builtin_amdgcn_s_wait_asynccnt(0);
#else
  asm volatile("s_wait_asynccnt 0x0" ::: "memory");
#endif
}

// Stage NC_ codebook columns (all 256 k values) into LDS with padded stride.
__device__ __forceinline__ void stage_load(float* dst, const float* __restrict__ cq,
                                           int s, int tid) {
  const float* src_base = cq + (size_t)s * NC_ * DD_;
#pragma unroll
  for (int it = 0; it < (NC_ * DD_ / 4) / 128; ++it) {  // 16 float4 per thread
    const int i   = tid + it * 128;
    const int col = i >> 6;            // 64 float4 per column
    const int kk  = (i & 63) << 2;
    async_load_b128(src_base + col * DD_ + kk, dst + col * COLSTRIDE_ + kk);
  }
}

__global__ __launch_bounds__(128) void rvq_main(const float* __restrict__ x,
                                                const float* __restrict__ cb,
                                                const float* __restrict__ cbsq,
                                                float* __restrict__ out) {
  __shared__ float ldsB[2][NC_ * COLSTRIDE_];  // ~33.8 KB, double buffered
  __shared__ int   ldsIdx[4 * 16];
  __shared__ float ldsRed[4];

  const int tid  = threadIdx.x;
  const int wave = tid >> 5;
  const int lane = tid & 31;
  const int rl   = lane & 15;   // row-in-strip / column-in-tile
  const int hi   = lane >> 4;   // which K-pair half of the A/B fragment

  const int row = blockIdx.x * ROWS_PER_BLOCK_ + wave * 16 + rl;  // my A row
  const int bbi = row / TT_;
  const int tti = row % TT_;
  const size_t xbase = (size_t)bbi * (DD_ * TT_) + tti;

  // Residual in A-fragment layout: a[j] holds r[row][4j+2*hi], r[row][4j+2*hi+1]
  v2f a[64];
#pragma unroll
  for (int j = 0; j < 64; ++j) {
    const int k0 = 4 * j + 2 * hi;
    a[j].x = x[xbase + (size_t)k0 * TT_];
    a[j].y = x[xbase + (size_t)(k0 + 1) * TT_];
  }

  float lossacc = 0.0f;

  for (int q = 0; q < NQ_; ++q) {
    const float* cq = cb + (size_t)q * KBINS_ * DD_;
    const float* sq = cbsq + q * KBINS_;

    float bestv[8];
    int   besti[8];
#pragma unroll
    for (int i = 0; i < 8; ++i) { bestv[i] = -3.4e38f; besti[i] = 0; }

    stage_load(ldsB[0], cq, 0, tid);  // prefetch stage 0

    for (int s = 0; s < NSTAGE_; ++s) {
      wait_asynccnt0();
      __syncthreads();  // my tile landed + everyone done with the other buffer
      if (s + 1 < NSTAGE_) stage_load(ldsB[(s + 1) & 1], cq, s + 1, tid);

      const float* bt = ldsB[s & 1];
#pragma unroll
      for (int sub = 0; sub < 2; ++sub) {
        v8f acc = {};  // C starts at 0
#pragma unroll
        for (int j = 0; j < 64; ++j) {
          const float2 bv =
              *(const float2*)&bt[(sub * 16 + rl) * COLSTRIDE_ + 4 * j + 2 * hi];
          v2f b;
          b.x = bv.x;
          b.y = bv.y;
          acc = __builtin_amdgcn_wmma_f32_16x16x4_f32(
              false, a[j], false, b, (short)0, acc, false, false);
        }
        const int   col = s * NC_ + sub * 16 + rl;
        const float cn  = sq[col];
#pragma unroll
        for (int i = 0; i < 8; ++i) {
          const float sc = 2.0f * acc[i] - cn;   // argmax(2 r.c - |c|^2)
          if (sc > bestv[i]) { bestv[i] = sc; besti[i] = col; }
        }
      }
    }

    // Per-row argmax across the 16 lanes of each half (first-index tiebreak).
#pragma unroll
    for (int i = 0; i < 8; ++i) {
#pragma unroll
      for (int off = 1; off < 16; off <<= 1) {
        const float ov = __shfl_xor(bestv[i], off, 32);
        const int   oi = __shfl_xor(besti[i], off, 32);
        if (ov > bestv[i] || (ov == bestv[i] && oi < besti[i])) {
          bestv[i] = ov;
          besti[i] = oi;
        }
      }
    }
    if (lane == 0 || lane == 16) {
#pragma unroll
      for (int i = 0; i < 8; ++i) ldsIdx[wave * 16 + hi * 8 + i] = besti[i];
    }
    __syncthreads();
    const int myidx = ldsIdx[wave * 16 + rl];

    if (lane < 16)
      out[CODES_OFF_ + (size_t)q * NROWS_ + row] = (float)myidx;  // codes[q,b,t]

    // residual -= cb[idx];  loss_q uses mean(residual_after^2)
    const float* crow = cq + (size_t)myidx * DD_;
#pragma unroll
    for (int j = 0; j < 64; ++j) {
      const float2 c = *(const float2*)&crow[4 * j + 2 * hi];
      a[j].x -= c.x;
      a[j].y -= c.y;
      lossacc += a[j].x * a[j].x + a[j].y * a[j].y;
    }
  }

  // quantized = x - residual_final, scattered back to (B, D, T)
#pragma unroll
  for (int j = 0; j < 64; ++j) {
    const int    k0 = 4 * j + 2 * hi;
    const size_t i0 = xbase + (size_t)k0 * TT_;
    const size_t i1 = xbase + (size_t)(k0 + 1) * TT_;
    out[i0] = x[i0] - a[j].x;
    out[i1] = x[i1] - a[j].y;
  }

  // penalty = sum(residual_after^2 over all q) / (NQ * N * D)
#pragma unroll
  for (int off = 1; off < 32; off <<= 1) lossacc += __shfl_xor(lossacc, off, 32);
  if (lane == 0) ldsRed[wave] = lossacc;
  __syncthreads();
  if (tid == 0) {
    const float s = ldsRed[0] + ldsRed[1] + ldsRed[2] + ldsRed[3];
    atomicAdd(&out[PEN_OFF_], s * (1.0f / ((float)NQ_ * NROWS_ * DD_)));
  }
}

// Precompute per-bin squared norms into d_ws and zero the penalty slot.
__global__ void rvq_init(const float* __restrict__ cb, float* __restrict__ cbsq,
                         float* __restrict__ out) {
  const int i = blockIdx.x * blockDim.x + threadIdx.x;
  if (i < NQ_ * KBINS_) {
    const float* r = cb + (size_t)i * DD_;
    float s = 0.0f;
    for (int d = 0; d < DD_; ++d) s += r[d] * r[d];
    cbsq[i] = s;
  }
  if (i == 0) out[PEN_OFF_] = 0.0f;
}

extern "C" void kernel_launch(void* const* d_in, const int* in_sizes, int n_in,
                              void* d_out, int out_size, void* d_ws, size_t ws_size,
                              hipStream_t stream) {
  (void)in_sizes; (void)n_in; (void)out_size; (void)ws_size;
  const float* x  = (const float*)d_in[0];
  const float* cb = (const float*)d_in[1];
  float* out  = (float*)d_out;
  float* cbsq = (float*)d_ws;  // 8*1024 floats = 32 KB

  rvq_init<<<(NQ_ * KBINS_ + 255) / 256, 256, 0, stream>>>(cb, cbsq, out);
  rvq_main<<<NROWS_ / ROWS_PER_BLOCK_, 128, 0, stream>>>(x, cb, cbsq, out);
}